// GINModel_55027120996387
// MI455X (gfx1250) — compile-verified
//
#include <hip/hip_runtime.h>
#include <stdint.h>

#define DD  128   // feature dim
#define DD2 256   // hidden dim

typedef float v2f __attribute__((ext_vector_type(2)));
typedef float v8f __attribute__((ext_vector_type(8)));

// ---------------------------------------------------------------------------
// Generic fp32 WMMA GEMM over row-major A[N,K] @ B[K,NOUT].
// MODE 0: lin_in   : Out = A@B + bias            (A staged via async->LDS)
// MODE 1: layer mm1: A'=(1+eps)*A + A2 fused at staging (explicit loads);
//                    Out = relu( bn( A'@B + bias ) )
// MODE 2: layer mm2: Out += A@B + bias           (A staged via async->LDS)
// Block: 256 threads = 8 waves; 64 rows (4 M-tiles); wave owns 16-col tiles.
// ---------------------------------------------------------------------------
template<int K, int NOUT, int MODE>
__global__ __launch_bounds__(256) void gemm_wmma(
    const float* __restrict__ A,  const float* __restrict__ A2,
    const float* __restrict__ B,  const float* __restrict__ bias,
    const float* __restrict__ gamma, const float* __restrict__ beta,
    const float* __restrict__ rmean, const float* __restrict__ rvar,
    const float* __restrict__ epsp,
    float* __restrict__ Out, int n_rows)
{
  constexpr int LDSK = K + 4;                 // pad: 4-bank row stride -> no conflicts
  __shared__ float Asmem[64 * LDSK];
  const int tid  = threadIdx.x;
  const int row0 = blockIdx.x * 64;

  constexpr int VECS = 64 * K / 4;            // float4 elements to stage

  if (MODE == 1) {
    // Explicit staging with fused combine: u = (1+eps)*h + agg
    const float onePlusEps = 1.0f + epsp[0];
    for (int i = tid; i < VECS; i += 256) {
      int r = i / (K / 4);
      int c = (i % (K / 4)) * 4;
      int rr = row0 + r;
      if (rr >= n_rows) rr = n_rows - 1;      // clamp (padding rows never stored)
      float4 v  = *(const float4*)(A  + (size_t)rr * K + c);
      float4 gq = *(const float4*)(A2 + (size_t)rr * K + c);
      v.x = onePlusEps * v.x + gq.x;
      v.y = onePlusEps * v.y + gq.y;
      v.z = onePlusEps * v.z + gq.z;
      v.w = onePlusEps * v.w + gq.w;
      *(float4*)&Asmem[r * LDSK + c] = v;
    }
  } else {
    // CDNA5 async direct-to-LDS staging: no VGPR round trip, tracked by ASYNCcnt.
    for (int i = tid; i < VECS; i += 256) {
      int r = i / (K / 4);
      int c = (i % (K / 4)) * 4;
      int rr = row0 + r;
      if (rr >= n_rows) rr = n_rows - 1;
      const float* gp = A + (size_t)rr * K + c;               // 16B aligned
      unsigned lds_addr = (unsigned)(uintptr_t)&Asmem[r * LDSK + c]; // wg-relative LDS byte addr
      asm volatile("global_load_async_to_lds_b128 %0, %1, off"
                   :: "v"(lds_addr), "v"(gp) : "memory");
    }
    asm volatile("s_wait_asynccnt 0x0" ::: "memory");          // my writes landed in LDS
  }
  __syncthreads();                                             // publish to all waves

  const int wave  = tid >> 5;                 // wave32
  const int lane  = tid & 31;
  const int l15   = lane & 15;
  const int khi   = (lane & 16) ? 2 : 0;      // A/B VGPR K split: lanes 16-31 -> K+2
  const int rhalf = (lane & 16) ? 8 : 0;      // C/D row split

  for (int ct = wave; ct < NOUT / 16; ct += 8) {
    const int ncol = ct * 16 + l15;           // column owned by this lane (B and C/D)
    v8f acc[4] = {};                          // 4 M-tiles accumulated in parallel
    for (int kk = 0; kk < K; kk += 4) {
      const int kb = kk + khi;
      v2f b;                                  // B 4x16 fragment
      b.x = B[(size_t)kb       * NOUT + ncol];
      b.y = B[(size_t)(kb + 1) * NOUT + ncol];
#pragma unroll
      for (int m = 0; m < 4; ++m) {           // reuse B across 4 M-tiles
        const float* ap = &Asmem[(m * 16 + l15) * LDSK + kb];
        v2f a; a.x = ap[0]; a.y = ap[1];      // ds_load_b64, conflict-free
        acc[m] = __builtin_amdgcn_wmma_f32_16x16x4_f32(
            false, a, false, b, (short)0, acc[m], false, false);
      }
    }

    // Per-lane epilogue constants (each lane owns exactly one output column).
    const float cb = bias[ncol];
    float scale = 1.0f, shift = 0.0f;
    if (MODE == 1) {
      const float s = gamma[ncol] * rsqrtf(rvar[ncol] + 1e-5f);
      scale = s;
      shift = beta[ncol] + (cb - rmean[ncol]) * s;
    }

#pragma unroll
    for (int m = 0; m < 4; ++m) {
#pragma unroll
      for (int v = 0; v < 8; ++v) {
        const int r = row0 + m * 16 + rhalf + v;
        if (r < n_rows) {
          const size_t o = (size_t)r * NOUT + ncol;
          float val = acc[m][v];
          if (MODE == 0)       val += cb;
          else if (MODE == 1)  val = fmaxf(val * scale + shift, 0.0f);
          else                 val += cb + Out[o];   // residual
          Out[o] = val;
        }
      }
    }
  }
}

// agg[dst] += relu(h[src]) : one wave per edge, 4 floats per lane (wave32 * 4 = 128)
__global__ __launch_bounds__(256) void edge_scatter(
    const float* __restrict__ h, const int* __restrict__ src,
    const int* __restrict__ dst, float* __restrict__ agg, int nE)
{
  const int e = blockIdx.x * 8 + (threadIdx.x >> 5);
  if (e >= nE) return;
  const int lane = threadIdx.x & 31;
  const int s = src[e], d = dst[e];
  const float4 v = *(const float4*)(h + (size_t)s * DD + lane * 4);
  float* o = agg + (size_t)d * DD + lane * 4;
  atomicAdd(o + 0, fmaxf(v.x, 0.0f));
  atomicAdd(o + 1, fmaxf(v.y, 0.0f));
  atomicAdd(o + 2, fmaxf(v.z, 0.0f));
  atomicAdd(o + 3, fmaxf(v.w, 0.0f));
}

// g[batch[node]] += h[node]
__global__ __launch_bounds__(256) void pool_scatter(
    const float* __restrict__ h, const int* __restrict__ batch,
    float* __restrict__ g, int n)
{
  const int i = blockIdx.x * blockDim.x + threadIdx.x;
  const int node = i >> 5;
  if (node >= n) return;
  const int lane = i & 31;
  const int b = batch[node];
  const float4 v = *(const float4*)(h + (size_t)node * DD + lane * 4);
  float* o = g + (size_t)b * DD + lane * 4;
  atomicAdd(o + 0, v.x);
  atomicAdd(o + 1, v.y);
  atomicAdd(o + 2, v.z);
  atomicAdd(o + 3, v.w);
}

__global__ void out_proj(const float* __restrict__ g, const float* __restrict__ w,
                         const float* __restrict__ b, float* __restrict__ out, int nG)
{
  const int gr = blockIdx.x * blockDim.x + threadIdx.x;
  if (gr >= nG) return;
  float s = 0.0f;
#pragma unroll 8
  for (int c = 0; c < DD; ++c) s += g[(size_t)gr * DD + c] * w[c];
  out[gr] = s + b[0];
}

__global__ void zero_f32(float* __restrict__ p, int n)
{
  const int i = (blockIdx.x * blockDim.x + threadIdx.x) * 4;
  if (i + 3 < n) {
    *(float4*)(p + i) = float4{0.0f, 0.0f, 0.0f, 0.0f};
  } else {
    for (int k = i; k < n; ++k) p[k] = 0.0f;
  }
}

extern "C" void kernel_launch(void* const* d_in, const int* in_sizes, int n_in,
                              void* d_out, int out_size, void* d_ws, size_t ws_size,
                              hipStream_t stream)
{
  const float* x        = (const float*)d_in[0];
  const int*   edge     = (const int*)d_in[1];
  const int*   batch    = (const int*)d_in[2];
  const float* lin_in_w = (const float*)d_in[3];
  const float* lin_in_b = (const float*)d_in[4];
  const float* eps      = (const float*)d_in[5];
  const float* W1       = (const float*)d_in[6];
  const float* b1       = (const float*)d_in[7];
  const float* gamma    = (const float*)d_in[8];
  const float* beta     = (const float*)d_in[9];
  const float* rmean    = (const float*)d_in[10];
  const float* rvar     = (const float*)d_in[11];
  const float* W2       = (const float*)d_in[12];
  const float* b2       = (const float*)d_in[13];
  const float* low      = (const float*)d_in[14];
  const float* lob      = (const float*)d_in[15];

  const int n  = in_sizes[0] / DD;     // N nodes
  const int nE = in_sizes[1] / 2;      // E edges
  const int nL = in_sizes[5];          // layers
  const int nG = out_size;             // graphs
  const int* src = edge;
  const int* dst = edge + nE;

  // Workspace layout (fp32): h[N,D] | agg[N,D] | z[N,2D] | g[G,D]  (~205 MB)
  float* h   = (float*)d_ws;
  float* agg = h   + (size_t)n * DD;
  float* z   = agg + (size_t)n * DD;
  float* g   = z   + (size_t)n * DD2;

  const int gemm_blocks = (n + 63) / 64;

  // h = x @ Win + b
  gemm_wmma<DD, DD, 0><<<gemm_blocks, 256, 0, stream>>>(
      x, nullptr, lin_in_w, lin_in_b,
      nullptr, nullptr, nullptr, nullptr, nullptr, h, n);

  for (int l = 0; l < nL; ++l) {
    zero_f32<<<((n * DD / 4) + 255) / 256, 256, 0, stream>>>(agg, n * DD);
    edge_scatter<<<(nE + 7) / 8, 256, 0, stream>>>(h, src, dst, agg, nE);
    // z = relu(bn(((1+eps)h + agg) @ W1 + b1))
    gemm_wmma<DD, DD2, 1><<<gemm_blocks, 256, 0, stream>>>(
        h, agg, W1 + (size_t)l * DD * DD2, b1 + (size_t)l * DD2,
        gamma + (size_t)l * DD2, beta + (size_t)l * DD2,
        rmean + (size_t)l * DD2, rvar + (size_t)l * DD2,
        eps + l, z, n);
    // h += z @ W2 + b2
    gemm_wmma<DD2, DD, 2><<<gemm_blocks, 256, 0, stream>>>(
        z, nullptr, W2 + (size_t)l * DD2 * DD, b2 + (size_t)l * DD,
        nullptr, nullptr, nullptr, nullptr, nullptr, h, n);
  }

  zero_f32<<<((nG * DD / 4) + 255) / 256, 256, 0, stream>>>(g, nG * DD);
  pool_scatter<<<((n * 32) + 255) / 256, 256, 0, stream>>>(h, batch, g, n);
  out_proj<<<(nG + 255) / 256, 256, 0, stream>>>(g, low, lob, (float*)d_out, nG);
}